// CTCLayer_30013231464524
// MI455X (gfx1250) — compile-verified
//
#include <hip/hip_runtime.h>
#include <math.h>

// ---------------------------------------------------------------------------
// CTC loss + greedy decode + normalized edit distance for MI455X (gfx1250).
// B=32, T=256, C=4000, L=32, S=2L+1=65, blank=C-1.
//
// Roofline: logits = 131 MB -> ~5.6us @ 23.3 TB/s. Kernel 1 streams each row
// twice (max+argmax, then sum of __expf(x-M)); the second pass hits L2 (the
// whole tensor fits in the 192 MB L2), keeping HBM traffic at 131 MB and the
// per-element VALU cost at ~5 ops so the pass stays bandwidth-bound.
// The 32-lane sum-reduce runs on the matrix pipe via two chained
// V_WMMA_F32_16X16X4_F32 (exact f32). Kernels 2/3 are L2-resident gather +
// per-batch barrier-free wave32 DP (CTC forward, greedy decode, Levenshtein
// via prefix-min shuffle scan).
// ---------------------------------------------------------------------------

typedef float v2f __attribute__((ext_vector_type(2)));
typedef float v8f __attribute__((ext_vector_type(8)));

#define Bsz   32
#define Tlen  256
#define Cnum  4000
#define Llen  32
#define Sext  65
#define LPSTR 66          // padded row stride for lp table (even -> float2 ok)
#define BLANK 3999
#define NEGV  (-1e9f)
#define EPSV  (1e-7f)

__device__ __forceinline__ float lae(float a, float b) {
  // jnp.logaddexp with finite -1e9 sentinel: max + log1p(exp(min-max))
  float mx = fmaxf(a, b);
  float dd = fminf(a, b) - mx;       // <= 0 (can be ~-1e9 -> exp -> 0)
  return mx + log1pf(expf(dd));
}

// Full-wave32 sum-reduce-broadcast on the matrix pipe, exact f32.
// WMMA #1 (A = {v,0}, B = ones, C = 0):  D-vgpr j = v_j + v_{j+16}  (lanes 0-15)
//                                                 = v_{j+8} + v_{j+24} (lanes 16-31)
// u = sum of the 8 D vgprs -> lanes 0-15 hold q_lo, lanes 16-31 hold q_hi.
// WMMA #2 (A = {u,0}) -> every lane of D vgpr0 = q_lo + q_hi = total.
__device__ __forceinline__ float wave_sum_wmma(float v) {
  v2f a;    a.x = v;    a.y = 0.0f;
  v2f ones; ones.x = 1.0f; ones.y = 1.0f;
  v8f c = {};
  v8f d = __builtin_amdgcn_wmma_f32_16x16x4_f32(false, a, false, ones,
                                                (short)0, c, false, false);
  float u = d[0] + d[1] + d[2] + d[3] + d[4] + d[5] + d[6] + d[7];
  v2f a2; a2.x = u; a2.y = 0.0f;
  v8f d2 = __builtin_amdgcn_wmma_f32_16x16x4_f32(false, a2, false, ones,
                                                 (short)0, c, false, false);
  return d2[0]; // total broadcast to all 32 lanes
}

// ---------------------------------------------------------------------------
// Kernel 1: streaming pass. One wave per (b,t) row of 4000 floats.
// Pass A: row max + first-occurrence argmax (branchless, ~3 VALU/elem).
// Pass B: s = sum __expf(x - M) with global M (1 trans op/elem, L2 re-read).
// 8 waves / block, 1024 blocks -> 8192 rows exactly; EXEC is all-ones at the
// WMMA reduction as the ISA requires (tail divergence reconverges first).
// ---------------------------------------------------------------------------
__global__ __launch_bounds__(256) void k_rowstats(const float* __restrict__ logits,
                                                  float* __restrict__ logZ,
                                                  int* __restrict__ amax) {
  const int wave = threadIdx.x >> 5;
  const int lane = threadIdx.x & 31;
  const int row  = blockIdx.x * 8 + wave;                 // < 8192
  const float4* rp = (const float4*)(logits + (size_t)row * Cnum); // 1000 x float4

  // ---- pass A: max + argmax (first occurrence) ---------------------------
  float m  = -INFINITY;
  int   bi = 0;
  for (int i = 0; i < 31; ++i) {
    float4 v = rp[lane + 32 * i];
    __builtin_prefetch((const void*)(rp + lane + 32 * (i + 6)), 0, 3);
    const int base = (lane + 32 * i) * 4;
    float xs[4] = {v.x, v.y, v.z, v.w};
#pragma unroll
    for (int k = 0; k < 4; ++k) {
      float x = xs[k];
      bi = (x > m) ? (base + k) : bi;       // in-order -> keeps first max
      m  = fmaxf(m, x);
    }
  }
  if (lane < 8) {                            // tail: float4 chunks 992..999
    float4 v = rp[992 + lane];
    const int base = (992 + lane) * 4;
    float xs[4] = {v.x, v.y, v.z, v.w};
#pragma unroll
    for (int k = 0; k < 4; ++k) {
      float x = xs[k];
      bi = (x > m) ? (base + k) : bi;
      m  = fmaxf(m, x);
    }
  }

  // cross-lane max + first-occurrence argmax (butterfly)
  float M = m; int I = bi;
#pragma unroll
  for (int off = 16; off >= 1; off >>= 1) {
    float ov = __shfl_xor(M, off);
    int   oi = __shfl_xor(I, off);
    if (ov > M || (ov == M && oi < I)) { M = ov; I = oi; }
  }

  // ---- pass B: sum of exp(x - M), row is L2/WGP$-resident ----------------
  float s = 0.0f;
  for (int i = 0; i < 31; ++i) {
    float4 v = rp[lane + 32 * i];
    s += __expf(v.x - M) + __expf(v.y - M) + __expf(v.z - M) + __expf(v.w - M);
  }
  if (lane < 8) {
    float4 v = rp[992 + lane];
    s += __expf(v.x - M) + __expf(v.y - M) + __expf(v.z - M) + __expf(v.w - M);
  }

  // wave sum on the matrix pipe (two chained f32 WMMAs)
  float tot = wave_sum_wmma(s);

  if (lane == 0) {
    logZ[row] = M + logf(tot);
    amax[row] = I;
  }
}

// ---------------------------------------------------------------------------
// Kernel 2: gather extended-label log-probs into a padded [row][66] table.
// lp(row,s) = log(exp(logit[row][ext_s] - logZ[row]) + eps). L2-resident.
// ---------------------------------------------------------------------------
__global__ __launch_bounds__(256) void k_gather_lp(const float* __restrict__ logits,
                                                   const int* __restrict__ labels,
                                                   const float* __restrict__ logZ,
                                                   float* __restrict__ lp) {
  const int tid   = blockIdx.x * 256 + threadIdx.x;
  const int total = Bsz * Tlen * Sext;
  if (tid >= total) return;
  const int row = tid / Sext;
  const int s   = tid - row * Sext;
  const int b   = row >> 8;                                // row / Tlen
  const int c   = (s & 1) ? labels[b * Llen + (s >> 1)] : BLANK;
  const float x = logits[(size_t)row * Cnum + c];
  lp[(size_t)row * LPSTR + s] = logf(expf(x - logZ[row]) + EPSV);
}

// ---------------------------------------------------------------------------
// Kernel 3: per-batch wave32 (no barriers inside loops).
//   (a) CTC forward DP: lane l holds alpha[2l], alpha[2l+1]; lane 31 also
//       alpha[64]. One shfl_up per timestep supplies alpha[s-1]/alpha[s-2].
//   (b) greedy decode: 8 preds/lane, wave prefix-sum compaction into LDS.
//   (c) Levenshtein: row recurrence via prefix-min scan over 32 columns.
// ---------------------------------------------------------------------------
__global__ __launch_bounds__(32) void k_ctc_dp(const int* __restrict__ labels,
                                               const int* __restrict__ input_length,
                                               const int* __restrict__ label_length,
                                               const float* __restrict__ lp,
                                               const int* __restrict__ amax,
                                               float* __restrict__ out) {
  __shared__ int dec[Tlen];
  const int b    = blockIdx.x;
  const int lane = threadIdx.x;            // 0..31
  const int len  = input_length[b];
  const int ll   = label_length[b];

  const int lab     = labels[b * Llen + lane];
  const int labprev = __shfl_up(lab, 1);
  const bool can_skip = (lane >= 1) && (lab != labprev);  // for state s = 2*lane+1

  const float* lpb = lp + (size_t)b * Tlen * LPSTR;

  // ---- (a) CTC forward DP -------------------------------------------------
  float2 lp0 = *(const float2*)(lpb + 2 * lane);
  float a0 = (lane == 0) ? lp0.x : NEGV;   // alpha[2l]
  float a1 = (lane == 0) ? lp0.y : NEGV;   // alpha[2l+1]
  float a2 = NEGV;                         // alpha[64] (lane 31 only meaningful)

  for (int t = 1; t < Tlen; ++t) {
    float2 lpv  = *(const float2*)(lpb + t * LPSTR + 2 * lane);
    float  lp64 = lpb[t * LPSTR + 64];
    __builtin_prefetch((const void*)(lpb + (t + 6) * LPSTR + 2 * lane), 0, 3);

    float prev0 = __shfl_up(a1, 1);        // alpha[2l-1]
    if (lane == 0) prev0 = NEGV;

    float n0 = lae(a0, prev0) + lpv.x;                               // even (blank)
    float n1 = lae(lae(a1, a0), can_skip ? prev0 : NEGV) + lpv.y;    // odd (label)
    float n2 = lae(a2, a1) + lp64;                                   // s = 64

    if (t < len) { a0 = n0; a1 = n1; a2 = n2; }    // else carry alpha (masked step)
  }

  float a_last = (ll <= 31) ? __shfl(a0, ll) : __shfl(a2, 31);  // alpha[2*ll]
  float a_prev = __shfl(a1, ll - 1);                            // alpha[2*ll-1]
  if (lane == 0) out[b] = -lae(a_last, a_prev);

  // ---- (b) greedy decode (collapse repeats, drop blanks) ------------------
  const int* pr = amax + b * Tlen;
  int p[8];
#pragma unroll
  for (int k = 0; k < 8; ++k) p[k] = pr[lane * 8 + k];
  int prevv = __shfl_up(p[7], 1);
  if (lane == 0) prevv = -1;

  int kv[8]; int cnt = 0;
#pragma unroll
  for (int k = 0; k < 8; ++k) {
    int t = lane * 8 + k;
    bool kp = (t < len) && (p[k] != BLANK) && (p[k] != prevv);
    if (kp) kv[cnt++] = p[k];
    prevv = p[k];
  }
  int inc = cnt;                            // inclusive prefix sum of counts
#pragma unroll
  for (int sh = 1; sh < 32; sh <<= 1) {
    int o = __shfl_up(inc, sh);
    if (lane >= sh) inc += o;
  }
  const int base = inc - cnt;
  const int n    = __shfl(inc, 31);         // hyp_len
  for (int j = 0; j < cnt; ++j) dec[base + j] = kv[j];
  __syncthreads();

  // ---- (c) Levenshtein: D[n][m] / m ---------------------------------------
  // Lane c <-> truth column j = c+1 (m = ll <= 32).
  // d[j] = min(best[j], d[j-1]+1), d[0] = i
  //      = j + min(i, prefix-min_{k<=j}(best[k]-k))     (5-step shfl min-scan)
  const int m  = ll;
  const int tv = lab;                       // truth symbol for this column
  int prow = lane + 1;                      // D[0][j] = j
  for (int i = 1; i <= n; ++i) {
    int h  = dec[i - 1];
    int dg = __shfl_up(prow, 1);            // D[i-1][j-1]
    if (lane == 0) dg = i - 1;
    int best = min(prow + 1, dg + ((h != tv) ? 1 : 0));
    int g = best - (lane + 1);
#pragma unroll
    for (int sh = 1; sh < 32; sh <<= 1) {
      int og = __shfl_up(g, sh);
      if (lane >= sh) g = min(g, og);
    }
    prow = (lane + 1) + min(g, i);
  }
  int Dnm = __shfl(prow, m - 1);
  if (lane == 0) out[Bsz + b] = (float)Dnm / (float)m;
}

// ---------------------------------------------------------------------------
// Host launch. Inputs: labels i32[32,32], logits f32[32,256,4000],
// input_length i32[32], label_length i32[32]. Output: f32[64] = loss | ed.
// Workspace: logZ (32KB) | argmax (32KB) | lp table (8192*66*4 = 2.07MB).
// ---------------------------------------------------------------------------
extern "C" void kernel_launch(void* const* d_in, const int* in_sizes, int n_in,
                              void* d_out, int out_size, void* d_ws, size_t ws_size,
                              hipStream_t stream) {
  const int*   labels = (const int*)d_in[0];
  const float* logits = (const float*)d_in[1];
  const int*   ilen   = (const int*)d_in[2];
  const int*   llen   = (const int*)d_in[3];
  float* out = (float*)d_out;

  char*  ws   = (char*)d_ws;
  float* logZ = (float*)ws;                         // 8192 floats
  int*   amax = (int*)(ws + 32768);                 // 8192 ints
  float* lptb = (float*)(ws + 65536);               // 8192 * 66 floats

  k_rowstats<<<Bsz * Tlen / 8, 256, 0, stream>>>(logits, logZ, amax);

  const int total = Bsz * Tlen * Sext;
  k_gather_lp<<<(total + 255) / 256, 256, 0, stream>>>(logits, labels, logZ, lptb);

  k_ctc_dp<<<Bsz, 32, 0, stream>>>(labels, ilen, llen, lptb, amax, out);
}